// TriangleUpdate_64321430225395
// MI455X (gfx1250) — compile-verified
//
#include <hip/hip_runtime.h>
#include <hip/hip_bf16.h>

typedef __attribute__((ext_vector_type(2))) float v2f;
typedef __attribute__((ext_vector_type(4))) float v4f;
typedef __attribute__((ext_vector_type(8))) float v8f;

#define BB 2
#define LL 512
#define DIM 128
#define KK 32
#define ROWS (BB * LL)   // 1024

// ---------------------------------------------------------------------------
// Kernel 1: K-sampled means.
//   grid.y==0: out_mean[b,i,d] = mean_k pair[b, i, idx[k], d]
//   grid.y==1: in_mean [b,j,d] = mean_k pair[b, idx[k], j, d]
// One 128-thread block per output row; fully coalesced 512B row loads.
// ---------------------------------------------------------------------------
__global__ __launch_bounds__(DIM) void means_kernel(
    const float* __restrict__ pair,
    float* __restrict__ out_mean,
    float* __restrict__ in_mean)
{
  const int r  = blockIdx.x;         // b*L + (i or j)
  const int b  = r >> 9;
  const int ij = r & (LL - 1);
  const int d  = threadIdx.x;
  float s = 0.0f;
  if (blockIdx.y == 0) {
    const float* base = pair + (size_t)r * LL * DIM + d;
#pragma unroll 4
    for (int k = 0; k < KK; ++k) {
      const int jk = (k * (LL - 1)) / (KK - 1);      // exact floor(linspace)
      s += base[(size_t)jk * DIM];
    }
    out_mean[(size_t)r * DIM + d] = s * (1.0f / KK);
  } else {
    const float* base = pair + ((size_t)b * LL) * LL * DIM + (size_t)ij * DIM + d;
#pragma unroll 4
    for (int k = 0; k < KK; ++k) {
      const int ik = (k * (LL - 1)) / (KK - 1);
      s += base[(size_t)ik * LL * DIM];
    }
    in_mean[(size_t)r * DIM + d] = s * (1.0f / KK);
  }
}

// ---------------------------------------------------------------------------
// Kernel 2: C = A(1024x128) @ W(128x128) + bias, fp32 via V_WMMA_F32_16X16X4_F32.
// Block = 256 threads = 8 waves; each wave owns one 16x16 tile (wave -> N-tile,
// block -> M-tile). 32 WMMAs per tile over K=128.
//   A frag (16x4): lanes 0-15 hold {K=k0,k0+1}, lanes 16-31 hold {k0+2,k0+3}.
//   B frag (4x16): V0 = rows k0 / k0+2 (lane halves), V1 = rows k0+1 / k0+3.
//   C (16x16): VGPR r -> M=r (lanes 0-15) / M=r+8 (lanes 16-31), N = lane&15.
// grid.y selects which (A,W,bias,C) quadruple -> two GEMMs per launch.
// bias==nullptr means no bias.
// ---------------------------------------------------------------------------
__global__ __launch_bounds__(256) void wmma_gemm_bias_kernel(
    const float* __restrict__ A0, const float* __restrict__ W0,
    const float* __restrict__ bias0, float* __restrict__ C0,
    const float* __restrict__ A1, const float* __restrict__ W1,
    const float* __restrict__ bias1, float* __restrict__ C1)
{
  const float* A;  const float* W;  const float* bias;  float* C;
  if (blockIdx.y == 0) { A = A0; W = W0; bias = bias0; C = C0; }
  else                 { A = A1; W = W1; bias = bias1; C = C1; }

  const int wave   = threadIdx.x >> 5;
  const int lane   = threadIdx.x & 31;
  const int half   = lane >> 4;        // 0: lanes 0-15, 1: lanes 16-31
  const int l16    = lane & 15;
  const int m_base = blockIdx.x * 16;
  const int n_base = wave * 16;

  v8f acc = {};
#pragma unroll 8
  for (int k0 = 0; k0 < DIM; k0 += 4) {
    const float* arow = A + (size_t)(m_base + l16) * DIM + (k0 + half * 2);
    v2f a;
    a.x = arow[0];
    a.y = arow[1];
    const float* wrow = W + (size_t)(k0 + half * 2) * DIM + (n_base + l16);
    v2f bfrag;
    bfrag.x = wrow[0];
    bfrag.y = wrow[DIM];
    acc = __builtin_amdgcn_wmma_f32_16x16x4_f32(
        /*neg_a=*/false, a, /*neg_b=*/false, bfrag,
        /*c_mod=*/(short)0, acc, /*reuse_a=*/false, /*reuse_b=*/false);
  }

  const float bn = bias ? bias[n_base + l16] : 0.0f;
#pragma unroll
  for (int r = 0; r < 8; ++r) {
    const int m = m_base + r + half * 8;
    C[(size_t)m * DIM + (n_base + l16)] = acc[r] + bn;
  }
}

// ---------------------------------------------------------------------------
// Kernel 3: fused gate + update + layernorm, streaming over (B, L, L, D).
// One block per (b,i) row-strip; 8 waves loop over j. Each wave handles one
// full 128-float row per j: lane owns 4 consecutive d's (b128 loads/stores).
// pair_repr is read NT and output written NT so the 512 MB stream does not
// evict the ~1 MB of hot (in_agg, gi) tables from L2 (192 MB, but NT keeps
// the replacement pressure off). Wave32 shfl_xor reduction for mean/var.
// ---------------------------------------------------------------------------
__global__ __launch_bounds__(256) void fused_gate_ln_kernel(
    const float* __restrict__ pair,
    const float* __restrict__ out_agg, const float* __restrict__ in_agg,
    const float* __restrict__ go,      const float* __restrict__ gi,
    const float* __restrict__ ln_w,    const float* __restrict__ ln_b,
    float* __restrict__ out)
{
  const int r    = blockIdx.x;        // b*L + i
  const int b    = r >> 9;
  const int wave = threadIdx.x >> 5;
  const int lane = threadIdx.x & 31;
  const int d0   = lane * 4;

  // Per-(b,i) constants for this block (held in registers across the j loop).
  const v4f oa  = *(const v4f*)(out_agg + (size_t)r * DIM + d0);
  const v4f gov = *(const v4f*)(go      + (size_t)r * DIM + d0);
  const v4f lw  = *(const v4f*)(ln_w + d0);
  const v4f lb  = *(const v4f*)(ln_b + d0);

  const float* prow = pair + (size_t)r * LL * DIM;
  float*       orow = out  + (size_t)r * LL * DIM;
  const float* iag  = in_agg + (size_t)b * LL * DIM;
  const float* girow = gi    + (size_t)b * LL * DIM;

  for (int j = wave; j < LL; j += 8) {
    const v4f ia  = *(const v4f*)(iag   + (size_t)j * DIM + d0);
    const v4f giv = *(const v4f*)(girow + (size_t)j * DIM + d0);
    const v4f pv  = __builtin_nontemporal_load(
        (const v4f*)(prow + (size_t)j * DIM + d0));

    v4f x;
    float s = 0.0f, s2 = 0.0f;
#pragma unroll
    for (int c = 0; c < 4; ++c) {
      const float logit = gov[c] + giv[c];                 // b_g folded into go
      const float g     = 1.0f / (1.0f + __expf(-logit));
      const float upd   = oa[c] + ia[c];
      const float xv    = pv[c] + g * (upd - pv[c]);
      x[c] = xv;
      s  += xv;
      s2 += xv * xv;
    }
    // wave32 reduction over the 128-wide row
#pragma unroll
    for (int off = 16; off > 0; off >>= 1) {
      s  += __shfl_xor(s,  off, 32);
      s2 += __shfl_xor(s2, off, 32);
    }
    const float mu   = s * (1.0f / DIM);
    const float var  = s2 * (1.0f / DIM) - mu * mu;
    const float rstd = rsqrtf(var + 1e-5f);

    v4f y;
#pragma unroll
    for (int c = 0; c < 4; ++c)
      y[c] = (x[c] - mu) * rstd * lw[c] + lb[c];

    __builtin_nontemporal_store(y, (v4f*)(orow + (size_t)j * DIM + d0));
  }
}

// ---------------------------------------------------------------------------
extern "C" void kernel_launch(void* const* d_in, const int* in_sizes, int n_in,
                              void* d_out, int out_size, void* d_ws, size_t ws_size,
                              hipStream_t stream) {
  const float* pair  = (const float*)d_in[0];
  const float* W_out = (const float*)d_in[1];
  const float* b_out = (const float*)d_in[2];
  const float* W_in  = (const float*)d_in[3];
  const float* b_in  = (const float*)d_in[4];
  const float* W_g   = (const float*)d_in[5];   // (2*DIM, DIM): Wg_o then Wg_i
  const float* b_g   = (const float*)d_in[6];
  const float* ln_w  = (const float*)d_in[7];
  const float* ln_b  = (const float*)d_in[8];
  float* out = (float*)d_out;

  const size_t NTAB = (size_t)ROWS * DIM;       // 131072 floats = 512 KB
  float* ws = (float*)d_ws;
  float* out_mean = ws + 0 * NTAB;
  float* in_mean  = ws + 1 * NTAB;
  float* out_agg  = ws + 2 * NTAB;
  float* in_agg   = ws + 3 * NTAB;
  float* go       = ws + 4 * NTAB;
  float* gi       = ws + 5 * NTAB;

  // Phase 1: sampled means (both directions in one launch).
  means_kernel<<<dim3(ROWS, 2), DIM, 0, stream>>>(pair, out_mean, in_mean);

  // Phase 2: out_agg = out_mean@W_out + b_out ; in_agg = in_mean@W_in + b_in
  wmma_gemm_bias_kernel<<<dim3(ROWS / 16, 2), 256, 0, stream>>>(
      out_mean, W_out, b_out, out_agg,
      in_mean,  W_in,  b_in,  in_agg);

  // Phase 3: go = out_agg@Wg_o + b_g ; gi = in_agg@Wg_i (no bias, avoid double add)
  wmma_gemm_bias_kernel<<<dim3(ROWS / 16, 2), 256, 0, stream>>>(
      out_agg, W_g,              b_g,     go,
      in_agg,  W_g + DIM * DIM,  nullptr, gi);

  // Phase 4: fused gate/update/layernorm streaming pass (bandwidth bound).
  fused_gate_ln_kernel<<<dim3(ROWS), 256, 0, stream>>>(
      pair, out_agg, in_agg, go, gi, ln_w, ln_b, out);
}